// Attention_55035710930956
// MI455X (gfx1250) — compile-verified
//
#include <hip/hip_runtime.h>
#include <math.h>
#include <stdint.h>

// CDNA5 (gfx1250) wave32 WMMA fp32 path: V_WMMA_F32_16X16X4_F32
typedef __attribute__((ext_vector_type(2))) float v2f;
typedef __attribute__((ext_vector_type(8))) float v8f;
typedef __attribute__((ext_vector_type(4))) unsigned int v4u;
typedef __attribute__((ext_vector_type(4))) int v4i;
typedef __attribute__((ext_vector_type(8))) int v8i;

__device__ __forceinline__ v8f wmma4(v2f a, v2f b, v8f c) {
  // 8 args: (neg_a, A, neg_b, B, c_mod, C, reuse_a, reuse_b)
  return __builtin_amdgcn_wmma_f32_16x16x4_f32(false, a, false, b, (short)0, c,
                                               false, false);
}

// ---------------------------------------------------------------------------
// Tensor Data Mover: async DMA of a 2-D fp32 tile (tile_w x tile_h) from
// global (row stride row_stride elements) into LDS at byte offset lds_off,
// optionally inserting pad DWORDs into LDS every 2^(pad_icode+1) DWORDs
// (D# pad feature) so fragment reads stay bank-conflict-free.
// Descriptor packing per CDNA5 ISA §8.3 (group0) / §8.4 (group1); groups 2/3
// zero (2-D tile: tile_dim2/3 = 0 -> unused). Tracked by TENSORcnt.
// This toolchain's builtin takes 6 args:
//   (uint32x4 g0, int32x8 g1, int32x4, int32x4, int32x8, i32 cpol)
// ---------------------------------------------------------------------------
__device__ __forceinline__ void tdm_load_2d_f32(
    unsigned lds_off, const float* gaddr, unsigned tile_w, unsigned tile_h,
    unsigned row_stride, unsigned tens_w, unsigned tens_h, unsigned pad_icode,
    unsigned pad_acode, bool pad_en) {
  const unsigned long long ga = (unsigned long long)(uintptr_t)gaddr;
  v4u g0;
  g0.x = 1u;                                   // count=1 (valid), user mode
  g0.y = lds_off;                              // LDS byte address
  g0.z = (unsigned)(ga & 0xFFFFFFFFu);         // global_addr[31:0]
  g0.w = (unsigned)((ga >> 32) & 0x01FFFFFFu)  // global_addr[56:32]
         | (2u << 30);                         // type = 2 ("image")
  v8i g1;
  g1[0] = (int)((2u << 16)                      // data_size = 4 bytes
                | (pad_en ? (1u << 20) : 0u)    // pad_enable
                | (pad_icode << 22)             // pad_interval code
                | (pad_acode << 25));           // pad_amount code (=DW-1)
  g1[1] = (int)((tens_w & 0xFFFFu) << 16);      // tensor_dim0[15:0]
  g1[2] = (int)(((tens_w >> 16) & 0xFFFFu) | ((tens_h & 0xFFFFu) << 16));
  g1[3] = (int)(((tens_h >> 16) & 0xFFFFu) | ((tile_w & 0xFFFFu) << 16));
  g1[4] = (int)(tile_h & 0xFFFFu);              // tile_dim1; tile_dim2 = 0
  g1[5] = (int)row_stride;                      // tensor_dim0_stride[31:0]
  g1[6] = 0;
  g1[7] = 0;
  const v4i z4 = {0, 0, 0, 0};
  const v8i z8 = {0, 0, 0, 0, 0, 0, 0, 0};
  __builtin_amdgcn_tensor_load_to_lds(g0, g1, z4, z4, z8, 0);
}

__device__ __forceinline__ unsigned lds_off_of(const void* p) {
  // LDS aperture: flat addr[31:0] == workgroup-relative LDS byte offset.
  return (unsigned)(uintptr_t)p;
}

// ---------------------------------------------------------------------------
// GEMM: Cout[M,N] = A[M,K] @ Bm[K,N] + bias[N]  (row-major fp32).
// Block = 256 threads = 8 waves, tile 128(M) x 64(N). K chunks of 32 are
// fetched by the TDM into double-buffered LDS panels while the previous chunk
// is consumed (wave 0 issues, s_wait_tensorcnt + barrier publishes):
//   lA[buf][row][k]  stride 36 (TDM pads 4 DW every 32) -> b64 A-frags
//   lB[buf][k][n]    stride 72 (TDM pads 8 DW every 64) -> paired b32 B-frags
//     (2*72 mod 64 == 16: the two lane-halves hit disjoint bank windows)
// Requires M%128==0, N%64==0, K%32==0.
// ---------------------------------------------------------------------------
__global__ __launch_bounds__(256) void gemm_bias_wmma(
    const float* __restrict__ A, const float* __restrict__ Bm,
    const float* __restrict__ bias, float* __restrict__ Cout,
    int M, int N, int K) {
  const int tid = threadIdx.x;
  const int lane = tid & 31;
  const int wave = tid >> 5;
  const int ml = lane & 15;
  const int hl = lane >> 4;
  const int koff = hl << 1;

  const int nb = blockIdx.x << 6;  // N block base
  const int mb = blockIdx.y << 7;  // M block base

  __shared__ float lA[2][128 * 36];
  __shared__ float lB[2][32 * 72];

  v8f acc0 = {}, acc1 = {}, acc2 = {}, acc3 = {};

  // Issue TDM loads for K-chunk kb into buffer buf (wave 0 only).
  auto issue = [&](int kb, int buf) {
    if (wave == 0) {
      tdm_load_2d_f32(lds_off_of(&lA[buf][0]), A + (long)mb * K + kb,
                      /*tile*/ 32, 128, /*stride*/ K, /*tens*/ K - kb, M - mb,
                      /*pad 4 DW every 32*/ 4, 3, true);
      tdm_load_2d_f32(lds_off_of(&lB[buf][0]), Bm + (long)kb * N + nb,
                      /*tile*/ 64, 32, /*stride*/ N, /*tens*/ N - nb, K - kb,
                      /*pad 8 DW every 64*/ 5, 7, true);
    }
  };

  issue(0, 0);  // prologue: prefetch first chunk

  const int rw = (wave << 4) + ml;  // this lane's A row within block tile
  for (int kb = 0, it = 0; kb < K; kb += 32, ++it) {
    const int buf = it & 1;
    const bool hasNext = (kb + 32) < K;
    if (hasNext) issue(kb + 32, buf ^ 1);  // overlap DMA with compute
    if (wave == 0) {
      if (hasNext)
        __builtin_amdgcn_s_wait_tensorcnt(2);  // current pair complete
      else
        __builtin_amdgcn_s_wait_tensorcnt(0);
    }
    __syncthreads();  // publish chunk `buf` to all waves

    const float* la = &lA[buf][0];
    const float* lb = &lB[buf][0];
#pragma unroll
    for (int k = 0; k < 32; k += 4) {
      const int kr = k + koff;
      const v2f a = *(const v2f*)(&la[rw * 36 + kr]);
      const v2f b0 = {lb[kr * 72 + ml + 0], lb[(kr + 1) * 72 + ml + 0]};
      const v2f b1 = {lb[kr * 72 + ml + 16], lb[(kr + 1) * 72 + ml + 16]};
      const v2f b2 = {lb[kr * 72 + ml + 32], lb[(kr + 1) * 72 + ml + 32]};
      const v2f b3 = {lb[kr * 72 + ml + 48], lb[(kr + 1) * 72 + ml + 48]};
      acc0 = wmma4(a, b0, acc0);
      acc1 = wmma4(a, b1, acc1);
      acc2 = wmma4(a, b2, acc2);
      acc3 = wmma4(a, b3, acc3);
    }
    __syncthreads();  // all reads of `buf` done before TDM refills it
  }

  const int n0 = nb + ml;
  const float bv0 = bias[n0], bv1 = bias[n0 + 16];
  const float bv2 = bias[n0 + 32], bv3 = bias[n0 + 48];
#pragma unroll
  for (int r = 0; r < 8; ++r) {
    const long row = (long)mb + (wave << 4) + r + (hl << 3);
    float* Crow = Cout + row * N + n0;
    Crow[0] = acc0[r] + bv0;
    Crow[16] = acc1[r] + bv1;
    Crow[32] = acc2[r] + bv2;
    Crow[48] = acc3[r] + bv3;
  }
}

// ---------------------------------------------------------------------------
// RMS-norm (over Dh) + RoPE, in place on the q and k slices of the qkv buffer.
// One 32-lane wave per (row, head, q|k); blockDim=64 => 2 independent waves.
// Lane owns elements {2*lane, 2*lane+1} = one RoPE pair (i = lane).
// ---------------------------------------------------------------------------
__global__ __launch_bounds__(64) void rmsnorm_rope(
    float* __restrict__ qkv, const float* __restrict__ cosb,
    const float* __restrict__ sinb, const float* __restrict__ qn_w,
    const float* __restrict__ kn_w, int N, int H, int Dh, int C3) {
  const int lane = threadIdx.x & 31;
  const int sel = threadIdx.x >> 5;  // 0 = q, 1 = k
  const int row = blockIdx.x / H;    // b*N + n
  const int h = blockIdx.x % H;
  const int n = row % N;
  const int C = C3 / 3;

  float* p = qkv + (long)row * C3 + sel * C + h * Dh + 2 * lane;
  v2f t = *(v2f*)p;
  float ss = t.x * t.x + t.y * t.y;
#pragma unroll
  for (int m = 16; m >= 1; m >>= 1) ss += __shfl_xor(ss, m, 32);
  const float inv = rsqrtf(ss * (1.0f / (float)Dh) + 1e-6f);

  const float* w = sel ? kn_w : qn_w;
  const float t0 = t.x * inv * w[2 * lane];
  const float t1 = t.y * inv * w[2 * lane + 1];
  const int half = Dh >> 1;
  const float c = cosb[(long)n * half + lane];
  const float s = sinb[(long)n * half + lane];
  v2f o = {t0 * c - t1 * s, t0 * s + t1 * c};
  *(v2f*)p = o;
}

// ---------------------------------------------------------------------------
// Flash attention (no mask), fp32 WMMA, Dh = 64 assumed.
// Block = 128 threads = 4 waves; wave w handles q-tile (blockIdx.x*4 + w) of
// one (b,h). 16-key K/V tiles are double-buffered in LDS and fetched by the
// TDM (wave 0 issues, tensorcnt + barrier publishes):
//   lK[buf][key][d] stride 68 -> conflict-free b64 K-fragment reads
//   lV[buf][key][d] stride 72 -> paired b32 V-fragment reads (disjoint banks)
//   pt[wave][q][key] stride 18: per-wave P bounce, C-layout -> A-layout
// Online softmax per 16-lane row group.
// ---------------------------------------------------------------------------
__global__ __launch_bounds__(128) void flash_attn_wmma(
    const float* __restrict__ qkv, float* __restrict__ out, int N, int H,
    int Dh) {
  const int tid = threadIdx.x;
  const int lane = tid & 31;
  const int wave = tid >> 5;
  const int ml = lane & 15;
  const int hl = lane >> 4;
  const int koff = hl << 1;
  const int C = H * Dh;
  const int C3 = 3 * C;
  const int qb = (blockIdx.x << 2) + wave;
  const int b = blockIdx.y / H;
  const int h = blockIdx.y % H;
  const float scale = rsqrtf((float)Dh);
  const long bbase = (long)b * N;

  __shared__ float lK[2][16 * 68];
  __shared__ float lV[2][16 * 72];
  __shared__ float pt[4][16 * 18];

  auto issueKV = [&](int jb, int buf) {
    if (wave == 0) {
      const float* kg = qkv + (bbase + jb) * C3 + C + h * Dh;
      tdm_load_2d_f32(lds_off_of(&lK[buf][0]), kg, 64, 16, C3, 64, 16,
                      /*pad 4 DW every 64*/ 5, 3, true);
      tdm_load_2d_f32(lds_off_of(&lV[buf][0]), kg + C, 64, 16, C3, 64, 16,
                      /*pad 8 DW every 64*/ 5, 7, true);
    }
  };

  // Preload Q as 16 A-fragments (K = 0..63 step 4), pre-scaled by Dh^-1/2.
  const float* qp = qkv + (bbase + (qb << 4) + ml) * C3 + h * Dh;
  v2f qf[16];
#pragma unroll
  for (int t = 0; t < 16; ++t) {
    v2f a = *(const v2f*)(qp + 4 * t + koff);
    qf[t].x = a.x * scale;
    qf[t].y = a.y * scale;
  }

  v8f o0 = {}, o1 = {}, o2 = {}, o3 = {};
  float rmax[8], rsum[8];
#pragma unroll
  for (int r = 0; r < 8; ++r) {
    rmax[r] = -1e30f;
    rsum[r] = 0.0f;
  }

  issueKV(0, 0);  // prologue prefetch
  for (int jb = 0, it = 0; jb < N; jb += 16, ++it) {
    const int buf = it & 1;
    const bool hasNext = (jb + 16) < N;
    if (hasNext) issueKV(jb + 16, buf ^ 1);
    if (wave == 0) {
      if (hasNext)
        __builtin_amdgcn_s_wait_tensorcnt(2);  // current pair complete
      else
        __builtin_amdgcn_s_wait_tensorcnt(0);
    }
    __syncthreads();

    // ---- S(16x16) = Q K^T ----
    const float* lk = &lK[buf][0];
    v8f s = {};
#pragma unroll
    for (int t = 0; t < 16; ++t) {
      const v2f bf = *(const v2f*)(&lk[ml * 68 + 4 * t + koff]);
      s = wmma4(qf[t], bf, s);
    }

    // ---- online softmax (rows live in 16-lane half-wave groups) ----
    float alpha[8];
#pragma unroll
    for (int r = 0; r < 8; ++r) {
      float m = s[r];
#pragma unroll
      for (int mm = 8; mm >= 1; mm >>= 1) m = fmaxf(m, __shfl_xor(m, mm, 32));
      const float nm = fmaxf(rmax[r], m);
      const float p = __expf(s[r] - nm);
      float ps = p;
#pragma unroll
      for (int mm = 8; mm >= 1; mm >>= 1) ps += __shfl_xor(ps, mm, 32);
      alpha[r] = __expf(rmax[r] - nm);
      rsum[r] = rsum[r] * alpha[r] + ps;
      rmax[r] = nm;
      pt[wave][(r + (hl << 3)) * 18 + ml] = p;  // same-wave DS is in-order
    }

#pragma unroll
    for (int r = 0; r < 8; ++r) {
      o0[r] *= alpha[r];
      o1[r] *= alpha[r];
      o2[r] *= alpha[r];
      o3[r] *= alpha[r];
    }

    // ---- O(16x64) += P(16x16) V(16x64) ----
    const float* lv = &lV[buf][0];
#pragma unroll
    for (int t = 0; t < 4; ++t) {
      const int kr = 4 * t + koff;
      const v2f a = *(const v2f*)(&pt[wave][ml * 18 + kr]);
      const v2f b0 = {lv[kr * 72 + ml + 0], lv[(kr + 1) * 72 + ml + 0]};
      const v2f b1 = {lv[kr * 72 + ml + 16], lv[(kr + 1) * 72 + ml + 16]};
      const v2f b2 = {lv[kr * 72 + ml + 32], lv[(kr + 1) * 72 + ml + 32]};
      const v2f b3 = {lv[kr * 72 + ml + 48], lv[(kr + 1) * 72 + ml + 48]};
      o0 = wmma4(a, b0, o0);
      o1 = wmma4(a, b1, o1);
      o2 = wmma4(a, b2, o2);
      o3 = wmma4(a, b3, o3);
    }
    __syncthreads();  // all reads of `buf` done before TDM refills it
  }

  // ---- normalize and store to (B,N,C) ----
#pragma unroll
  for (int r = 0; r < 8; ++r) {
    const float invs = 1.0f / rsum[r];
    const long row = bbase + (qb << 4) + r + (hl << 3);
    float* op = out + row * C + h * Dh + ml;
    op[0] = o0[r] * invs;
    op[16] = o1[r] * invs;
    op[32] = o2[r] * invs;
    op[48] = o3[r] * invs;
  }
}

// ---------------------------------------------------------------------------
extern "C" void kernel_launch(void* const* d_in, const int* in_sizes, int n_in,
                              void* d_out, int out_size, void* d_ws,
                              size_t ws_size, hipStream_t stream) {
  const float* x = (const float*)d_in[0];
  const float* cosb = (const float*)d_in[1];
  const float* sinb = (const float*)d_in[2];
  const float* qkv_w = (const float*)d_in[3];
  const float* qkv_b = (const float*)d_in[4];
  const float* proj_w = (const float*)d_in[5];
  const float* proj_b = (const float*)d_in[6];
  const float* qn_w = (const float*)d_in[7];
  const float* kn_w = (const float*)d_in[8];

  // Derive dims (B=2, N=2048, C=768, H=12, Dh=64 for this task).
  const int C3 = in_sizes[4];            // 3*C
  const int C = C3 / 3;
  const int Dh = in_sizes[7];            // qn_w
  const int H = C / Dh;
  const int M = in_sizes[0] / C;         // B*N
  const int N = in_sizes[1] / (Dh / 2);  // cos is (N, Dh/2)
  const int B = M / N;

  float* ws_qkv = (float*)d_ws;             // M x 3C
  float* ws_att = ws_qkv + (size_t)M * C3;  // M x C

  // 1) qkv = x @ qkv_w + qkv_b
  gemm_bias_wmma<<<dim3(C3 / 64, M / 128), 256, 0, stream>>>(x, qkv_w, qkv_b,
                                                             ws_qkv, M, C3, C);
  // 2) RMS-norm + RoPE on q,k slices in place
  rmsnorm_rope<<<M * H, 64, 0, stream>>>(ws_qkv, cosb, sinb, qn_w, kn_w, N, H,
                                         Dh, C3);
  // 3) fused flash attention -> (B,N,C)
  flash_attn_wmma<<<dim3(N / 64, B * H), 128, 0, stream>>>(ws_qkv, ws_att, N,
                                                           H, Dh);
  // 4) out = att @ proj_w + proj_b
  gemm_bias_wmma<<<dim3(C / 64, M / 128), 256, 0, stream>>>(
      ws_att, proj_w, proj_b, (float*)d_out, M, C, C);
}